// cheb_conv_withSAt_71408126263413
// MI455X (gfx1250) — compile-verified
//
#include <hip/hip_runtime.h>
#include <hip/hip_bf16.h>

typedef float v2f __attribute__((ext_vector_type(2)));
typedef float v8f __attribute__((ext_vector_type(8)));

static constexpr int Bn = 64, Nn = 22, Fin = 64, Fout = 64, Tn = 512, Kk = 3;
static constexpr int TT  = 16;          // t-tile width per workgroup
static constexpr int KN  = Kk * Nn;     // 66 combined (k,n) contraction size
static constexpr int KNP = 68;          // padded to multiple of 4
static constexpr int XST = 66;          // Xs stride over f (padded, even)
static constexpr int TST = 66;          // Ts stride over f (padded, even)
static constexpr int GST = 68;          // Gs stride over kn (even)
static constexpr int YST = 74;          // Ys stride over kn (padded, even, conflict-free)

// Dynamic LDS layout (floats)
static constexpr int XS_OFF = 0;
static constexpr int XS_SZ  = Nn * TT * XST;     // 23232 : Xs[(n*16+tt)*66 + f]
static constexpr int TS_OFF = XS_OFF + XS_SZ;
static constexpr int TS_SZ  = Kk * Fout * TST;   // 12672 : Ts[(k*64+o)*66 + f] = theta[k][f][o]
static constexpr int GS_OFF = TS_OFF + TS_SZ;
static constexpr int GS_SZ  = 32 * GST;          //  2176 : Gs[m*68 + kn]
static constexpr int YS_OFF = GS_OFF + GS_SZ;
static constexpr int YS_SZ  = 256 * YST;         // 18944 : Ys[(olocal*16+tt)*74 + kn]
static constexpr int SMEM_FLOATS = YS_OFF + YS_SZ;   // 57024 floats = 228096 B (< 320 KB/WGP)

// ---------------------------------------------------------------------------
// Prep 1: supports = softmax(relu(E E^T)); S0=I, S1=P, S2=2*P@P - I  -> ws
// ---------------------------------------------------------------------------
__global__ __launch_bounds__(512) void precompute_S(const float* __restrict__ emb,
                                                    float* __restrict__ S) {
  __shared__ float Es[Nn * Nn], Ps[Nn * Nn], S2s[Nn * Nn];
  const int tid = threadIdx.x;
  if (tid < Nn * Nn) Es[tid] = emb[tid];
  __syncthreads();
  const int i = tid / Nn, j = tid % Nn;
  if (tid < Nn * Nn) {
    float d = 0.f;
    for (int l = 0; l < Nn; ++l) d += Es[i * Nn + l] * Es[j * Nn + l];
    Ps[tid] = fmaxf(d, 0.f);
  }
  __syncthreads();
  if (tid < Nn) {  // row-wise softmax (axis=1), one thread per row
    float mx = -1e30f;
    for (int c = 0; c < Nn; ++c) mx = fmaxf(mx, Ps[tid * Nn + c]);
    float s = 0.f;
    for (int c = 0; c < Nn; ++c) s += expf(Ps[tid * Nn + c] - mx);
    const float inv = 1.0f / s;
    for (int c = 0; c < Nn; ++c) Ps[tid * Nn + c] = expf(Ps[tid * Nn + c] - mx) * inv;
  }
  __syncthreads();
  if (tid < Nn * Nn) {
    float d = 0.f;
    for (int l = 0; l < Nn; ++l) d += Ps[i * Nn + l] * Ps[l * Nn + j];
    S2s[tid] = 2.f * d - (i == j ? 1.f : 0.f);
  }
  __syncthreads();
  if (tid < Nn * Nn) {
    S[tid]               = (i == j) ? 1.f : 0.f;
    S[Nn * Nn + tid]     = Ps[tid];
    S[2 * Nn * Nn + tid] = S2s[tid];
  }
}

// ---------------------------------------------------------------------------
// Prep 2: Gt[b][m][kn] = S[k][n][m] * att[b][n][m], zero-padded to 32x68
// ---------------------------------------------------------------------------
__global__ __launch_bounds__(256) void precompute_G(const float* __restrict__ S,
                                                    const float* __restrict__ att,
                                                    float* __restrict__ Gt) {
  const int b = blockIdx.x;
  for (int idx = threadIdx.x; idx < 32 * GST; idx += 256) {
    const int m = idx / GST, j = idx % GST;
    float v = 0.f;
    if (m < Nn && j < KN) {
      const int k = j / Nn, n = j - k * Nn;
      v = S[(k * Nn + n) * Nn + m] * att[((size_t)b * Nn + n) * Nn + m];
    }
    Gt[(size_t)b * GS_SZ + idx] = v;
  }
}

// ---------------------------------------------------------------------------
// Main fused kernel: one workgroup per (b, 16-wide t tile), 512 threads.
// Stage A (WMMA): Y[k,n,o16,tt16] = theta_k^T @ x_b[n]   (K = f = 64)
// Stage B (WMMA): out[m,o,tt]     = G_b @ Y              (K = kn = 68 padded)
// X tile staged via ASYNCcnt-tracked global->LDS copies (transpose in flight).
// ---------------------------------------------------------------------------
__global__ __launch_bounds__(512, 1) void cheb_main(const float* __restrict__ x,
                                                    const float* __restrict__ theta,
                                                    const float* __restrict__ Gt,
                                                    float* __restrict__ out) {
  extern __shared__ float sm[];
  float* Xs = sm + XS_OFF;
  float* Ts = sm + TS_OFF;
  float* Gs = sm + GS_OFF;
  float* Ys = sm + YS_OFF;

  const int tid  = threadIdx.x;
  const int lane = tid & 31;
  const int wave = tid >> 5;           // 0..15
  const int b    = blockIdx.y;
  const int t0   = blockIdx.x * TT;
  const int ln   = lane & 15;          // row/col index within a 16-tile
  const int joff = (lane >> 4) << 1;   // K pair select: 0 (lanes 0-15) / 2 (lanes 16-31)
  const int hi   = (lane >> 4) * 8;    // C/D row offset for high half-wave

  // Async-stage the x tile global->LDS, transposing to f-contiguous layout:
  // Xs[(n*16+tt)*66 + f] = x[b][n][f][t0+tt].  Per-lane LDS dest addresses let
  // GLOBAL_LOAD_ASYNC_TO_LDS_B32 do the transpose in flight (tracked: ASYNCcnt).
  for (int idx = tid; idx < Nn * Fin * TT; idx += 512) {
    const int n  = idx >> 10;          // /(Fin*TT)
    const int r  = idx & 1023;
    const int f  = r >> 4;
    const int tt = r & 15;
    const uint32_t lds_addr =
        (uint32_t)(size_t)&Xs[(n * TT + tt) * XST + f];   // flat->LDS offset (addr[31:0])
    const float* gaddr = &x[(((size_t)b * Nn + n) * Fin + f) * Tn + t0 + tt];
    asm volatile("global_load_async_to_lds_b32 %0, %1, off"
                 :: "v"(lds_addr), "v"(gaddr)
                 : "memory");
  }
  // Stage in theta transposed: Ts[(k*64+o)*66 + f] = theta[k][f][o] (coalesced on o)
  for (int idx = tid; idx < Kk * Fin * Fout; idx += 512) {
    const int o = idx & 63;
    const int f = (idx >> 6) & 63;
    const int k = idx >> 12;
    Ts[(k * Fout + o) * TST + f] = theta[(k * Fin + f) * Fout + o];
  }
  // Stage in G_b
  for (int idx = tid; idx < GS_SZ; idx += 512) Gs[idx] = Gt[(size_t)b * GS_SZ + idx];
  // Zero Y's kn padding columns (66,67); never rewritten afterwards
  {
    const int c = tid >> 1, j = KN + (tid & 1);
    Ys[c * YST + j] = 0.0f;
  }
  asm volatile("s_wait_asynccnt 0" ::: "memory");  // own-wave async copies done
  __syncthreads();

  for (int ot = 0; ot < Fout / 16; ++ot) {   // 4 o-tiles
    // ---- Stage A: 66 (k,n) tiles of 16(o) x 16(tt), K=f=64 -> 16 WMMA each ----
    for (int q = wave; q < KN; q += 16) {
      const int k = q / Nn, n = q - k * Nn;
      v8f acc = {};
      const float* ta = &Ts[(k * Fout + ot * 16 + ln) * TST + joff];  // A: [o, f]
      const float* xb = &Xs[(n * TT + ln) * XST + joff];              // B: [f, tt]
#pragma unroll
      for (int j = 0; j < Fin / 4; ++j) {
        const v2f a  = *reinterpret_cast<const v2f*>(ta + 4 * j);
        const v2f bb = *reinterpret_cast<const v2f*>(xb + 4 * j);
        acc = __builtin_amdgcn_wmma_f32_16x16x4_f32(false, a, false, bb,
                                                    (short)0, acc, false, false);
      }
#pragma unroll
      for (int i = 0; i < 8; ++i)   // D tile -> Ys[(olocal*16+tt)*74 + kn]
        Ys[((i + hi) * 16 + ln) * YST + q] = acc[i];
    }
    __syncthreads();
    // ---- Stage B: 2 m-tiles x 16 c-tiles, K=kn=68 -> 17 WMMA each ----
    for (int p = wave; p < 32; p += 16) {
      const int mt = p >> 4, ct = p & 15;
      v8f acc = {};
      const float* ga = &Gs[(mt * 16 + ln) * GST + joff];        // A: [m, kn]
      const float* yb = &Ys[(ct * 16 + ln) * YST + joff];        // B: [kn, c]
#pragma unroll
      for (int j = 0; j < KNP / 4; ++j) {
        const v2f a  = *reinterpret_cast<const v2f*>(ga + 4 * j);
        const v2f bb = *reinterpret_cast<const v2f*>(yb + 4 * j);
        acc = __builtin_amdgcn_wmma_f32_16x16x4_f32(false, a, false, bb,
                                                    (short)0, acc, false, false);
      }
      const int o = ot * 16 + ct;   // each c-tile maps to one output feature o
#pragma unroll
      for (int i = 0; i < 8; ++i) {
        const int m = mt * 16 + i + hi;
        if (m < Nn)
          out[(((size_t)b * Nn + m) * Fout + o) * Tn + t0 + ln] = fmaxf(acc[i], 0.0f);
      }
    }
    __syncthreads();   // Ys reused next o-tile
  }
}

// ---------------------------------------------------------------------------
extern "C" void kernel_launch(void* const* d_in, const int* in_sizes, int n_in,
                              void* d_out, int out_size, void* d_ws, size_t ws_size,
                              hipStream_t stream) {
  const float* x     = (const float*)d_in[0];  // [64,22,64,512]
  const float* att   = (const float*)d_in[1];  // [64,22,22]
  const float* theta = (const float*)d_in[2];  // [3,64,64]
  const float* emb   = (const float*)d_in[3];  // [22,22]
  float* out = (float*)d_out;                  // [64,22,64,512]

  float* wsS = (float*)d_ws;                   // 3*22*22 floats
  float* wsG = wsS + 2048;                     // 64*32*68 floats

  precompute_S<<<1, 512, 0, stream>>>(emb, wsS);
  precompute_G<<<Bn, 256, 0, stream>>>(wsS, att, wsG);

  dim3 grid(Tn / TT, Bn);                      // 32 x 64 workgroups
  const size_t smem = (size_t)SMEM_FLOATS * sizeof(float);  // 228 KB dynamic LDS
  cheb_main<<<grid, 512, smem, stream>>>(x, theta, wsG, out);
}